// KNN_9242769621831
// MI455X (gfx1250) — compile-verified
//
#include <hip/hip_runtime.h>
#include <float.h>
#include <math.h>

typedef __attribute__((ext_vector_type(2))) float v2f;
typedef __attribute__((ext_vector_type(8))) float v8f;

static constexpr int KNN = 16;
static constexpr int BB  = 4;
static constexpr int DD  = 3;
static constexpr int RR  = 8192;
static constexpr int QQ  = 8192;
static constexpr int BLOCK = 128;                 // 4 waves of 32
static constexpr int WAVES_PER_BLOCK = BLOCK / 32;

// One wave handles a strip of 16 queries over all R refs.
// Distance tiles come from V_WMMA_F32_16X16X4_F32 with the K dim padded:
//   A(refs)  row = (rx, ry, rz, r2)
//   B(query) col = (qx, qy, qz, -0.5)
//   C        col = -0.5*q2
// => accumulator D = r.q - 0.5 r2 - 0.5 q2 = -0.5 * d2.
// Top-k keeps the 16 LARGEST accumulator values per query.
__global__ __launch_bounds__(BLOCK) void knn_wmma_kernel(
    const float* __restrict__ ref, const float* __restrict__ query,
    float* __restrict__ dist_out, int* __restrict__ idx_out)
{
  __shared__ float sKey[WAVES_PER_BLOCK][16][32];
  __shared__ int   sIdx[WAVES_PER_BLOCK][16][32];

  const int lane = threadIdx.x & 31;
  const int w    = threadIdx.x >> 5;
  const int waveGlobal     = blockIdx.x * WAVES_PER_BLOCK + w;
  const int stripsPerBatch = QQ / 16;
  const int b  = waveGlobal / stripsPerBatch;
  const int q0 = (waveGlobal % stripsPerBatch) * 16;

  const float* refB = ref   + (size_t)b * DD * RR;
  const float* qryB = query + (size_t)b * DD * QQ;

  const int m  = lane & 15;          // matrix row/col index within half-wave
  const int hi = (lane >> 4) & 1;    // upper half-wave => K=2,3 / M=8..15

  // ---- B operand: query columns (fixed for the whole strip) ----
  const int   q  = q0 + m;
  const float qx = qryB[0 * QQ + q];
  const float qy = qryB[1 * QQ + q];
  const float qz = qryB[2 * QQ + q];
  v2f bmat;
  bmat.x = hi ? qz : qx;             // K=0 / K=2
  bmat.y = hi ? -0.5f : qy;          // K=1 / K=3 (pad slot carries -0.5)

  // ---- C operand: -0.5*||q||^2 broadcast down the column ----
  const float qn2  = fmaf(qx, qx, fmaf(qy, qy, qz * qz));
  const float cval = -0.5f * qn2;
  v8f cmat;
  #pragma unroll
  for (int j = 0; j < 8; ++j) cmat[j] = cval;

  // ---- per-lane top-16 (register resident; replace-min policy) ----
  float bestK[KNN];
  int   bestI[KNN];
  #pragma unroll
  for (int i = 0; i < KNN; ++i) { bestK[i] = -FLT_MAX; bestI[i] = 0; }
  float wmin = -FLT_MAX;
  int   wpos = 0;

  for (int t = 0; t < RR / 16; ++t) {
    const int   r0 = t * 16;
    const float rx = refB[0 * RR + r0 + m];
    const float ry = refB[1 * RR + r0 + m];
    const float rz = refB[2 * RR + r0 + m];
    const float r2 = fmaf(rx, rx, fmaf(ry, ry, rz * rz));
    v2f amat;
    amat.x = hi ? rz : rx;           // K=0 / K=2
    amat.y = hi ? r2 : ry;           // K=1 / K=3 (pad slot carries ||r||^2)

    v8f acc = __builtin_amdgcn_wmma_f32_16x16x4_f32(
        false, amat, false, bmat, (short)0, cmat, false, false);

    const int rbase = r0 + (hi << 3);   // VGPR j holds row M = j + 8*hi
    #pragma unroll
    for (int j = 0; j < 8; ++j) {
      const float c = acc[j];
      if (c > wmin) {
        #pragma unroll
        for (int i = 0; i < KNN; ++i)
          if (i == wpos) { bestK[i] = c; bestI[i] = rbase + j; }
        float nm = bestK[0]; int np = 0;
        #pragma unroll
        for (int i = 1; i < KNN; ++i)
          if (bestK[i] < nm) { nm = bestK[i]; np = i; }
        wmin = nm; wpos = np;
      }
    }
  }

  // ---- merge the two half-lists of each query column via LDS ----
  #pragma unroll
  for (int i = 0; i < KNN; ++i) {
    sKey[w][m][hi * 16 + i] = bestK[i];
    sIdx[w][m][hi * 16 + i] = bestI[i];
  }
  __syncthreads();

  if (lane < 16) {
    // selection sort: 16 largest keys (= 16 smallest d2), descending key
    for (int k = 0; k < KNN; ++k) {
      int   bp = k;
      float bv = sKey[w][m][k];
      for (int j2 = k + 1; j2 < 32; ++j2) {
        const float v = sKey[w][m][j2];
        if (v > bv) { bv = v; bp = j2; }
      }
      const float tk = sKey[w][m][k];
      const int   ti = sIdx[w][m][k];
      sKey[w][m][k]  = sKey[w][m][bp];
      sIdx[w][m][k]  = sIdx[w][m][bp];
      sKey[w][m][bp] = tk;
      sIdx[w][m][bp] = ti;

      const float d2 = fmaxf(-2.0f * sKey[w][m][k], 0.0f);
      dist_out[((size_t)b * KNN + k) * QQ + q] = sqrtf(d2);
      idx_out [((size_t)b * KNN + k) * QQ + q] = sIdx[w][m][k];
    }
  }
}

extern "C" void kernel_launch(void* const* d_in, const int* in_sizes, int n_in,
                              void* d_out, int out_size, void* d_ws, size_t ws_size,
                              hipStream_t stream) {
  (void)in_sizes; (void)n_in; (void)out_size; (void)d_ws; (void)ws_size;
  const float* ref   = (const float*)d_in[0];   // [B, D, R]
  const float* query = (const float*)d_in[1];   // [B, D, Q]
  float* dist = (float*)d_out;                              // [B, K, Q] f32
  int*   idx  = (int*)(dist + (size_t)BB * KNN * QQ);       // [B, K, Q] i32

  const int totalWaves = BB * (QQ / 16);              // 2048 strips
  const int grid = totalWaves / WAVES_PER_BLOCK;      // 512 blocks
  knn_wmma_kernel<<<grid, BLOCK, 0, stream>>>(ref, query, dist, idx);
}